// PartialCRF_45423574122875
// MI455X (gfx1250) — compile-verified
//
#include <hip/hip_runtime.h>

typedef __attribute__((ext_vector_type(2))) float v2f;
typedef __attribute__((ext_vector_type(8))) float v8f;

#define CRF_IMPOSSIBLE (-10000000.0f)

namespace {
constexpr int kT = 2048;
constexpr int kK = 48;
}

// One wave (32 threads) handles 16 sequences for either the denominator
// (blockIdx.y==0) or the partially-masked numerator (blockIdx.y==1).
//
// State labeling permutation: register slot (mt, h, r) of the 16x16 f32
// C/D layout (lane = 16h+n, VGPR r; n = batch) holds CRF state
//   s(mt,h,r) = 16*mt + 4*(r>>1) + 2*h + (r&1)
// which makes the exp'd accumulator registers directly reusable as the
// B operand tiles of v_wmma_f32_16x16x4_f32 for the next step (B tile kk
// = comps 2*(kk&3), 2*(kk&3)+1 of accumulator kk>>2) -- no shuffles.
extern "C" __global__ void __launch_bounds__(32)
crf_partial_kernel(const float* __restrict__ em,
                   const int*   __restrict__ tags,
                   const int*   __restrict__ mask,
                   const float* __restrict__ startT,
                   const float* __restrict__ endT,
                   const float* __restrict__ trans,
                   float*       __restrict__ partial)
{
  const int lane = threadIdx.x & 31;
  const int n    = lane & 15;        // batch row within tile / A-matrix row
  const int h    = lane >> 4;        // lane half
  const int hb2  = h << 1;
  const int b    = blockIdx.x * 16 + n;
  const bool is_num = (blockIdx.y != 0);

  const size_t rowE = (size_t)b * kT * kK;  // emissions row base (elements)
  const size_t rowM = (size_t)b * kT;       // tags/mask row base

  // ---- constant A operand: AE[mt][kk] = exp(transitions)^T tiles --------
  // A is 16x4 f32: lane&15 = row m (output state j, permuted), cols are
  // input states i = 4*kk + 2*h + v (natural order thanks to permutation).
  v2f AE[3][12];
  {
    const int m    = n;
    const int joff = 4 * ((m & 7) >> 1) + 2 * (m >> 3) + (m & 1);
#pragma unroll
    for (int mt = 0; mt < 3; ++mt) {
      const int j = 16 * mt + joff;
#pragma unroll
      for (int kk = 0; kk < 12; ++kk) {
        const int i0 = 4 * kk + hb2;
        v2f a;
        a.x = __expf(trans[i0 * kK + j]);
        a.y = __expf(trans[(i0 + 1) * kK + j]);
        AE[mt][kk] = a;
      }
    }
  }

  // ---- sequence length (mask is 1..1 0..0 per row) -----------------------
  int len = 0;
  {
    const int4* mr = (const int4*)(mask + rowM);
#pragma unroll 4
    for (int i = 0; i < kT / 4; ++i) {
      int4 v = mr[i];
      len += v.x + v.y + v.z + v.w;
    }
  }

  // ---- alpha init: start + emissions[t=0] (numerator's pt[0] mask is
  //      deferred to the cur-mask of step 1 / the final et mask) ----------
  v8f alpha[3];
#pragma unroll
  for (int mt = 0; mt < 3; ++mt) {
#pragma unroll
    for (int q = 0; q < 4; ++q) {
      const int c = 16 * mt + 4 * q + hb2;
      float2 s = *(const float2*)(startT + c);
      float2 e = *(const float2*)(em + rowE + c);
      alpha[mt][2 * q]     = s.x + e.x;
      alpha[mt][2 * q + 1] = s.y + e.y;
    }
  }

  // ---- forward recursion -------------------------------------------------
#pragma unroll 1
  for (int t = 1; t < kT; ++t) {
    // issue this step's loads early so they overlap the alpha math
    float2 e[3][4];
    const float* emt = em + rowE + (size_t)t * kK + hb2;
#pragma unroll
    for (int mt = 0; mt < 3; ++mt)
#pragma unroll
      for (int q = 0; q < 4; ++q)
        e[mt][q] = *(const float2*)(emt + 16 * mt + 4 * q);
    int tg = 0;
    if (is_num) tg = tags[rowM + t - 1];
    if (t + 16 < kT)
      __builtin_prefetch(em + rowE + (size_t)(t + 16) * kK + h * 24, 0, 0);

    // per-sequence row max (rescale for exp-domain accumulation)
    float mx = alpha[0][0];
#pragma unroll
    for (int mt = 0; mt < 3; ++mt)
#pragma unroll
      for (int r = 0; r < 8; ++r)
        mx = fmaxf(mx, alpha[mt][r]);
    mx = fmaxf(mx, __shfl_xor(mx, 16));

    // B operand: exp(alpha - mx), numerator additionally masked by the
    // possible-tags of position t-1 (exp(IMPOSSIBLE) == 0)
    v8f vx[3];
#pragma unroll
    for (int mt = 0; mt < 3; ++mt)
#pragma unroll
      for (int r = 0; r < 8; ++r)
        vx[mt][r] = __expf(alpha[mt][r] - mx);
    if (is_num) {
#pragma unroll
      for (int mt = 0; mt < 3; ++mt)
#pragma unroll
        for (int r = 0; r < 8; ++r) {
          const int i_s = 16 * mt + 4 * (r >> 1) + (r & 1) + hb2;
          vx[mt][r] = (tg < 0 || tg == i_s) ? vx[mt][r] : 0.0f;
        }
    }

    // S^T = E^T * V^T : three independent 12-deep f32 WMMA chains (K=48)
    v8f acc[3];
#pragma unroll
    for (int mt = 0; mt < 3; ++mt) {
      v8f c = {0.f, 0.f, 0.f, 0.f, 0.f, 0.f, 0.f, 0.f};
#pragma unroll
      for (int kk = 0; kk < 12; ++kk) {
        v2f bt;
        bt.x = vx[kk >> 2][(kk & 3) * 2];
        bt.y = vx[kk >> 2][(kk & 3) * 2 + 1];
        c = __builtin_amdgcn_wmma_f32_16x16x4_f32(
                false, AE[mt][kk], false, bt, (short)0, c, false, false);
      }
      acc[mt] = c;
    }

    // alpha' = em + mx + log(S); carry old alpha past end of sequence
    const bool active = (t < len);
#pragma unroll
    for (int mt = 0; mt < 3; ++mt)
#pragma unroll
      for (int r = 0; r < 8; ++r) {
        const float ev = (r & 1) ? e[mt][r >> 1].y : e[mt][r >> 1].x;
        const float s  = fmaxf(acc[mt][r], 1e-37f);
        const float na = ev + mx + __logf(s);
        alpha[mt][r] = active ? na : alpha[mt][r];
      }
  }

  // ---- final logsumexp with end transitions ------------------------------
  float w[3][8];
#pragma unroll
  for (int mt = 0; mt < 3; ++mt)
#pragma unroll
    for (int q = 0; q < 4; ++q) {
      float2 ev = *(const float2*)(endT + 16 * mt + 4 * q + hb2);
      w[mt][2 * q]     = ev.x;
      w[mt][2 * q + 1] = ev.y;
    }
  if (is_num) {
    const int tl = tags[rowM + (len - 1)];
#pragma unroll
    for (int mt = 0; mt < 3; ++mt)
#pragma unroll
      for (int r = 0; r < 8; ++r) {
        const int j_s = 16 * mt + 4 * (r >> 1) + (r & 1) + hb2;
        if (!(tl < 0 || tl == j_s)) w[mt][r] = CRF_IMPOSSIBLE;
      }
  }
  float mx = alpha[0][0] + w[0][0];
#pragma unroll
  for (int mt = 0; mt < 3; ++mt)
#pragma unroll
    for (int r = 0; r < 8; ++r)
      mx = fmaxf(mx, alpha[mt][r] + w[mt][r]);
  mx = fmaxf(mx, __shfl_xor(mx, 16));
  float ssum = 0.f;
#pragma unroll
  for (int mt = 0; mt < 3; ++mt)
#pragma unroll
    for (int r = 0; r < 8; ++r)
      ssum += __expf(alpha[mt][r] + w[mt][r] - mx);
  ssum += __shfl_xor(ssum, 16);
  float res = mx + __logf(ssum);
  if (is_num) res = -res;

  // sum over the 16 sequences of this tile (halves are duplicates)
  res += __shfl_xor(res, 8);
  res += __shfl_xor(res, 4);
  res += __shfl_xor(res, 2);
  res += __shfl_xor(res, 1);
  if (lane == 0) partial[blockIdx.y * 16 + blockIdx.x] = res;
}

// Deterministic fixed-order reduction of the 32 block partials.
extern "C" __global__ void crf_reduce_kernel(const float* __restrict__ p,
                                             float* __restrict__ out) {
  if (threadIdx.x == 0) {
    float s = 0.f;
    for (int i = 0; i < 32; ++i) s += p[i];
    out[0] = s;
  }
}

extern "C" void kernel_launch(void* const* d_in, const int* in_sizes, int n_in,
                              void* d_out, int out_size, void* d_ws, size_t ws_size,
                              hipStream_t stream) {
  const float* em   = (const float*)d_in[0];
  const int*   tags = (const int*)d_in[1];
  const int*   mask = (const int*)d_in[2];
  const float* st   = (const float*)d_in[3];
  const float* en   = (const float*)d_in[4];
  const float* tr   = (const float*)d_in[5];
  float* partials   = (float*)d_ws;

  const int B = in_sizes[2] / kT;          // mask is [B, T]
  dim3 grid(B / 16, 2);
  crf_partial_kernel<<<grid, 32, 0, stream>>>(em, tags, mask, st, en, tr,
                                              partials);
  crf_reduce_kernel<<<1, 32, 0, stream>>>(partials, (float*)d_out);
}